// SelectiveWKV_V2_5368709120023
// MI455X (gfx1250) — compile-verified
//
#include <hip/hip_runtime.h>

// ---------------- CDNA5 (gfx1250) WMMA types ----------------
typedef __attribute__((ext_vector_type(2))) float v2f;
typedef __attribute__((ext_vector_type(8))) float v8f;

// D = A(16x4,f32) * B(4x16,f32) + C(16x16,f32), wave32.
#define WMMA_F32(a, b, c) \
  __builtin_amdgcn_wmma_f32_16x16x4_f32(false, (a), false, (b), (short)0, (c), false, false)

// CDNA5 async global->LDS copy (16B per lane), tracked by ASYNCcnt.
__device__ __forceinline__ void async_copy16(uint32_t lds_byte_off, const void* gptr) {
  asm volatile("global_load_async_to_lds_b128 %0, %1, off"
               :: "v"(lds_byte_off), "v"((uint64_t)(uintptr_t)gptr)
               : "memory");
}
__device__ __forceinline__ void async_wait0() {
  asm volatile("s_wait_asynccnt 0" ::: "memory");
}

// ---------------- problem dims ----------------
constexpr int Bb = 8, T = 2048, D = 768, H = 12, HS = 64;
constexpr int BT = Bb * T;   // 16384 rows
constexpr int NC = T / 64;   // 32 chunks of length 64

// ---------------- weight transpose: WT[d*768+e] = W[e*768+d] ----------------
__global__ __launch_bounds__(256) void transpose768(const float* __restrict__ W,
                                                    float* __restrict__ WT) {
  int i = blockIdx.x * 256 + threadIdx.x;   // 589824 elements
  int d = i / 768, e = i % 768;
  WT[i] = W[e * 768 + d];
}

// ---------------- LayerNorm (one row per block) ----------------
__global__ __launch_bounds__(256) void ln_kernel(const float* __restrict__ x,
                                                 const float* __restrict__ gamma,
                                                 const float* __restrict__ beta,
                                                 float* __restrict__ xn) {
  __shared__ float s1[256], s2[256];
  int row = blockIdx.x, tid = threadIdx.x;
  const float* xr = x + (size_t)row * D;
  float a0 = xr[tid], a1 = xr[tid + 256], a2 = xr[tid + 512];
  s1[tid] = a0 + a1 + a2;
  s2[tid] = a0 * a0 + a1 * a1 + a2 * a2;
  __syncthreads();
  for (int s = 128; s > 0; s >>= 1) {
    if (tid < s) { s1[tid] += s1[tid + s]; s2[tid] += s2[tid + s]; }
    __syncthreads();
  }
  float mu  = s1[0] * (1.0f / D);
  float var = s2[0] * (1.0f / D) - mu * mu;
  float inv = rsqrtf(var + 1e-5f);
  float* o = xn + (size_t)row * D;
  o[tid]       = (a0 - mu) * inv * gamma[tid]       + beta[tid];
  o[tid + 256] = (a1 - mu) * inv * gamma[tid + 256] + beta[tid + 256];
  o[tid + 512] = (a2 - mu) * inv * gamma[tid + 512] + beta[tid + 512];
}

// ---------------- WMMA GEMM: C(16384x768) = A(16384x768) * Bm(768x768) ----------------
// Each wave computes a 16(M) x 64(N) strip = 4 WMMA tiles, reusing the A
// fragment 4x per k-step (cuts L2 A-traffic 4x vs one-tile-per-wave).
// EPI 0: plain store (BT x D)
// EPI 1: decay = sigmoid(-(c+bias[col])) scattered to (B,H,T,HS)
// EPI 2: plain value scattered to (B,H,T,HS)
// EPI 3: sigmoid(c) stored (BT x D)
template <int EPI>
__global__ __launch_bounds__(256) void gemm_wmma_k(const float* __restrict__ A,
                                                   const float* __restrict__ Bm,
                                                   const float* __restrict__ bias,
                                                   float* __restrict__ Cout) {
  constexpr int N = D, K = D;
  constexpr int strips_n = N / 64;                 // 12
  int wid = blockIdx.x * 8 + (threadIdx.x >> 5);   // one 16x64 strip per wave
  int tm = wid / strips_n, ts = wid % strips_n;
  int l   = threadIdx.x & 31;
  int row = l & 15;
  int kp  = (l >> 4) << 1;                         // 0 or 2
  int hi8 = (l >> 4) << 3;

  const float* ap = A  + (size_t)(tm * 16 + row) * K + kp;
  const float* bp = Bm + (size_t)kp * N + ts * 64 + row;
  v8f cacc[4] = {};
#pragma unroll 2
  for (int kk = 0; kk < K / 4; ++kk) {
    v2f a; a.x = ap[0]; a.y = ap[1];
#pragma unroll
    for (int j = 0; j < 4; ++j) {
      v2f b; b.x = bp[j * 16]; b.y = bp[N + j * 16];
      cacc[j] = WMMA_F32(a, b, cacc[j]);
    }
    ap += 4;
    bp += 4 * (size_t)N;
  }
#pragma unroll
  for (int j = 0; j < 4; ++j) {
    int col = ts * 64 + j * 16 + row;
#pragma unroll
    for (int r = 0; r < 8; ++r) {
      int grow = tm * 16 + hi8 + r;   // global row (b*T + t)
      float val = cacc[j][r];
      if constexpr (EPI == 0) {
        Cout[(size_t)grow * N + col] = val;
      } else if constexpr (EPI == 1) {
        float z = val + bias[col];
        float dec = 1.0f / (1.0f + __expf(z));       // 1 - sigmoid(z)
        int b_ = grow >> 11, t_ = grow & 2047, h_ = col >> 6, hs_ = col & 63;
        Cout[(((size_t)(b_ * H + h_)) * T + t_) * HS + hs_] = dec;
      } else if constexpr (EPI == 2) {
        int b_ = grow >> 11, t_ = grow & 2047, h_ = col >> 6, hs_ = col & 63;
        Cout[(((size_t)(b_ * H + h_)) * T + t_) * HS + hs_] = val;
      } else {
        Cout[(size_t)grow * N + col] = 1.0f / (1.0f + __expf(-val));
      }
    }
  }
}

// ---------------- chunked selective scan (all matmuls via WMMA) ----------------
// Per (b,h): 32 sequential chunks of L=64. Within chunk (all 64x64):
//   P = column-wise cumulative product of decay d (Hillis-Steele, 6 steps)
//   Kc <- A = K .* P ;  Dc <- Bq = K .* rcp(P)      (division-free hot loop)
//   Y = A*S0 + tril(A*Bq^T)*V ;  S = diag(P_L) * (S0 + Bq^T*V)
// 16 waves = 4x4 tile grid of each 64x64 matmul. Chunk staging uses the
// CDNA5 async global->LDS DMA path (ASYNCcnt).
__global__ __launch_bounds__(512) void scan_kernel(const float* __restrict__ Kg,
                                                   const float* __restrict__ Vg,
                                                   const float* __restrict__ Dg,
                                                   const float* __restrict__ Rg,
                                                   float* __restrict__ Yg) {
  __shared__ __align__(16) float Kc[4096];  // K -> A -> masked G
  __shared__ __align__(16) float Vc[4096];  // V
  __shared__ __align__(16) float Dc[4096];  // decay -> P -> Bq
  __shared__ __align__(16) float Sm[4096];  // running state S (i x j)

  int bh = blockIdx.x;
  int b  = bh / H, h = bh % H;
  int tid = threadIdx.x, wave = tid >> 5, l = tid & 31;
  int tm = wave >> 2, tn = wave & 3;
  int row = l & 15;
  int kp  = (l >> 4) << 1;
  int hi8 = (l >> 4) << 3;
  int trow = tm * 16 + row;   // M-dim index of A-fragment rows
  int jcol = tn * 16 + row;   // N-dim index of B/C-fragment cols

  uint32_t lds_k = (uint32_t)(uintptr_t)(&Kc[0]);
  uint32_t lds_v = (uint32_t)(uintptr_t)(&Vc[0]);
  uint32_t lds_d = (uint32_t)(uintptr_t)(&Dc[0]);

  for (int i = tid; i < 4096; i += 512) Sm[i] = 0.0f;

  size_t base0 = (size_t)bh * T * HS;
  for (int c0 = 0; c0 < NC; ++c0) {
    __syncthreads();  // protect Kc/Vc/Dc reuse across chunks
    // --- async DMA of 64x64 K,V,decay chunk (contiguous in (B,H,T,HS)) ---
    {
      size_t gb = base0 + (size_t)c0 * 64 * HS;   // element offset, 16B aligned
      const char* kg = (const char*)(Kg + gb);
      const char* vg = (const char*)(Vg + gb);
      const char* dg = (const char*)(Dg + gb);
#pragma unroll
      for (int q = 0; q < 2; ++q) {
        uint32_t bo = (uint32_t)(tid + q * 512) * 16u;   // 1024 x 16B per array
        async_copy16(lds_k + bo, kg + bo);
        async_copy16(lds_v + bo, vg + bo);
        async_copy16(lds_d + bo, dg + bo);
      }
      async_wait0();
    }
    __syncthreads();

    // --- column-wise cumulative product Dc <- P (Hillis-Steele over t) ---
#pragma unroll
    for (int s = 1; s < 64; s <<= 1) {
      float tmp[8];
#pragma unroll
      for (int q = 0; q < 8; ++q) {
        int idx = tid + q * 512;
        int t = idx >> 6;
        float v = Dc[idx];
        if (t >= s) v *= Dc[idx - s * 64];
        tmp[q] = v;
      }
      __syncthreads();
#pragma unroll
      for (int q = 0; q < 8; ++q) Dc[tid + q * 512] = tmp[q];
      __syncthreads();
    }

    // snapshot P_L for this wave's state-tile rows before Dc is overwritten
    float pl[8];
#pragma unroll
    for (int r = 0; r < 8; ++r) pl[r] = Dc[63 * 64 + tm * 16 + hi8 + r];
    __syncthreads();

    // --- in-place: Kc <- A = K.*P ; Dc <- Bq = K.*rcp(P) (one rcp/elem) ---
#pragma unroll
    for (int q = 0; q < 8; ++q) {
      int i = tid + q * 512;
      float k = Kc[i], p = Dc[i];
      Kc[i] = k * p;
      Dc[i] = k * __builtin_amdgcn_rcpf(p);
    }
    __syncthreads();

    // --- three 64x64 matmuls (division-free) ---
    v8f y = {}, z = {}, g = {};
#pragma unroll 2
    for (int kk = 0; kk < 16; ++kk) {
      int k0 = kk * 4 + kp;
      // A-fragment rows (t): A[trow][k0], A[trow][k0+1]
      v2f a; a.x = Kc[trow * 64 + k0]; a.y = Kc[trow * 64 + k0 + 1];
      // Y1 = A * S0
      v2f bs; bs.x = Sm[k0 * 64 + jcol]; bs.y = Sm[(k0 + 1) * 64 + jcol];
      y = WMMA_F32(a, bs, y);
      // G = A * Bq^T  (only lower-triangular tiles are ever consumed)
      if (tm >= tn) {
        v2f bg; bg.x = Dc[jcol * 64 + k0]; bg.y = Dc[jcol * 64 + k0 + 1];
        g = WMMA_F32(a, bg, g);
      }
      // Z = Bq^T * V  (A-frag [m=i][k=s] = Bq[s][i])
      v2f az; az.x = Dc[k0 * 64 + trow]; az.y = Dc[(k0 + 1) * 64 + trow];
      v2f bv; bv.x = Vc[k0 * 64 + jcol]; bv.y = Vc[(k0 + 1) * 64 + jcol];
      z = WMMA_F32(az, bv, z);
    }
    __syncthreads();  // all reads of Kc(A), Dc(Bq), Sm(S0) complete

    // --- write masked G into Kc (lower tiles only); S = P_L .* (S0 + Z) ---
#pragma unroll
    for (int r = 0; r < 8; ++r) {
      int gr = tm * 16 + hi8 + r;   // t (for G) / i (for S,Z) element row
      if (tm >= tn) {
        float gv = (tm > tn || gr >= jcol) ? g[r] : 0.0f;
        Kc[gr * 64 + jcol] = gv;
      }
      Sm[gr * 64 + jcol] = pl[r] * (Sm[gr * 64 + jcol] + z[r]);
    }
    __syncthreads();  // G complete

    // --- Y += G * V  (G[t][s]=0 for s>t, so only kk < (tm+1)*4 contribute) ---
    const int kkmax = (tm + 1) * 4;
#pragma unroll 4
    for (int kk = 0; kk < kkmax; ++kk) {
      int k0 = kk * 4 + kp;
      v2f a;  a.x = Kc[trow * 64 + k0]; a.y = Kc[trow * 64 + k0 + 1];
      v2f bv; bv.x = Vc[k0 * 64 + jcol]; bv.y = Vc[(k0 + 1) * 64 + jcol];
      y = WMMA_F32(a, bv, y);
    }
    // --- fused y *= r, store to (B,T,D) ---
#pragma unroll
    for (int r = 0; r < 8; ++r) {
      int t_loc = tm * 16 + hi8 + r;
      size_t grow = (size_t)b * T + c0 * 64 + t_loc;
      int gcol = h * HS + jcol;
      size_t idx = grow * D + gcol;
      Yg[idx] = y[r] * Rg[idx];
    }
  }
}

// ---------------- host side ----------------
extern "C" void kernel_launch(void* const* d_in, const int* in_sizes, int n_in,
                              void* d_out, int out_size, void* d_ws, size_t ws_size,
                              hipStream_t stream) {
  const float* x     = (const float*)d_in[0];
  const float* Wx    = (const float*)d_in[1];
  const float* Ww    = (const float*)d_in[2];
  const float* bw    = (const float*)d_in[3];
  const float* Wk    = (const float*)d_in[4];
  const float* Wv    = (const float*)d_in[5];
  const float* Wr    = (const float*)d_in[6];
  const float* Wo    = (const float*)d_in[7];
  const float* gamma = (const float*)d_in[8];
  const float* beta  = (const float*)d_in[9];

  float* ws = (float*)d_ws;
  const size_t SZ  = (size_t)BT * D;   // 12,582,912 floats per activation tensor
  const size_t WSZ = (size_t)D * D;    // 589,824 floats per weight
  float* XN  = ws;
  float* XP  = ws + 1 * SZ;
  float* KS  = ws + 2 * SZ;
  float* VS  = ws + 3 * SZ;
  float* DSC = ws + 4 * SZ;            // decay (1-w) in scan layout
  float* RS  = ws + 5 * SZ;
  float* YB  = ws + 6 * SZ;
  float* WTb = ws + 7 * SZ;
  float* WxT = WTb + 0 * WSZ;
  float* WwT = WTb + 1 * WSZ;
  float* WkT = WTb + 2 * WSZ;
  float* WvT = WTb + 3 * WSZ;
  float* WrT = WTb + 4 * WSZ;
  float* WoT = WTb + 5 * WSZ;

  // weight transposes (Bmat[d][e] = W[e][d]) for coalesced WMMA B-loads
  transpose768<<<2304, 256, 0, stream>>>(Wx, WxT);
  transpose768<<<2304, 256, 0, stream>>>(Ww, WwT);
  transpose768<<<2304, 256, 0, stream>>>(Wk, WkT);
  transpose768<<<2304, 256, 0, stream>>>(Wv, WvT);
  transpose768<<<2304, 256, 0, stream>>>(Wr, WrT);
  transpose768<<<2304, 256, 0, stream>>>(Wo, WoT);

  ln_kernel<<<BT, 256, 0, stream>>>(x, gamma, beta, XN);

  const int GB = (BT / 16) * (D / 64) / 8;   // 1536 blocks, 8 strip-waves each
  gemm_wmma_k<0><<<GB, 256, 0, stream>>>(XN, WxT, nullptr, XP);    // x_proj
  gemm_wmma_k<1><<<GB, 256, 0, stream>>>(XP, WwT, bw, DSC);        // decay = 1-sigmoid(.+bw), scan layout
  gemm_wmma_k<2><<<GB, 256, 0, stream>>>(XN, WkT, nullptr, KS);    // k, scan layout
  gemm_wmma_k<2><<<GB, 256, 0, stream>>>(XN, WvT, nullptr, VS);    // v, scan layout
  gemm_wmma_k<3><<<GB, 256, 0, stream>>>(XN, WrT, nullptr, RS);    // r = sigmoid

  scan_kernel<<<Bb * H, 512, 0, stream>>>(KS, VS, DSC, RS, YB);    // y = scan(k,v,w) * r

  gemm_wmma_k<0><<<GB, 256, 0, stream>>>(YB, WoT, nullptr, (float*)d_out);
}